// Quantizer_62594853372449
// MI455X (gfx1250) — compile-verified
//
#include <hip/hip_runtime.h>
#include <hip/hip_bf16.h>
#include <math.h>

// ---------------------------------------------------------------------------
// Residual-VQ quantizer for MI455X (gfx1250, wave32).
//  - GEMMs via V_WMMA_F32_16X16X4_F32 (full fp32: integer argmax output is
//    tie-sensitive; bf16 WMMA could flip indexes).
//  - x tile staged into LDS by the Tensor Data Mover (tensor_load_to_lds),
//    with D# LDS-padding (2 DWORDs per 256) giving a 516-float row pitch so
//    16-lane A-fragment column reads hit distinct banks (4*r+c mod 64).
//  - Everything fits in L2 (x=1MB, W=4MB): stream W with HT prefetch.
//  - Subset-selection step reduced to an 8x8 Gram matrix (45 dots of len 512)
//    instead of the reference's (256x512) expansion.
// ---------------------------------------------------------------------------

#define DIM      512
#define CBS      256
#define NCB      8
#define BSZ      512
#define LDSPITCH 516   // 512 + 2x2 pad DWORDs (one 2-DW pad per 256 DW)
// element (r,d) lives at  r*516 + d + 2*(d>=256)   (matches TDM pad layout)

typedef float v2f __attribute__((ext_vector_type(2)));
typedef float v8f __attribute__((ext_vector_type(8)));
typedef unsigned int v4u __attribute__((ext_vector_type(4)));
typedef int v4i __attribute__((ext_vector_type(4)));
typedef int v8i __attribute__((ext_vector_type(8)));

#if defined(__has_builtin)
#if __has_builtin(__builtin_amdgcn_tensor_load_to_lds) && \
    __has_builtin(__builtin_amdgcn_s_wait_tensorcnt)
#define HAVE_TDM 1
#endif
#endif

__device__ __forceinline__ float wave_sum32(float v) {
#pragma unroll
  for (int o = 16; o > 0; o >>= 1) v += __shfl_down(v, o, 32);
  return v;
}

// Shared WMMA core: acc += As_tile(16xK) * W_rows(N=32 cols per wave)^T.
// As uses the split-pad layout; W rows are read as 8-byte B-fragments.
__device__ __forceinline__ void gemm_core_16x32(
    const float* __restrict__ As, const float* __restrict__ w0,
    const float* __restrict__ w1, int l16, int hf, v8f& acc0, v8f& acc1) {
#pragma unroll
  for (int kh = 0; kh < 2; ++kh) {
    const float* a0  = As + l16 * LDSPITCH + kh * 258 + 2 * hf;
    const float* wp0 = w0 + kh * 256;
    const float* wp1 = w1 + kh * 256;
#pragma unroll 4
    for (int k0 = 0; k0 < 256; k0 += 4) {
      if ((k0 & 60) == 0) {  // every 256B: high-temporal prefetch (W is hot)
        __builtin_prefetch(wp0 + k0 + 64, 0, 3);
        __builtin_prefetch(wp1 + k0 + 64, 0, 3);
      }
      v2f a;   a.x   = a0[k0];  a.y   = a0[k0 + 1];
      v2f bv0; bv0.x = wp0[k0]; bv0.y = wp0[k0 + 1];
      v2f bv1; bv1.x = wp1[k0]; bv1.y = wp1[k0 + 1];
      acc0 = __builtin_amdgcn_wmma_f32_16x16x4_f32(false, a, false, bv0,
                                                   (short)0, acc0, false, false);
      acc1 = __builtin_amdgcn_wmma_f32_16x16x4_f32(false, a, false, bv1,
                                                   (short)0, acc1, false, false);
    }
  }
}

// ---------------------------------------------------------------------------
// c2[c*CBS+k] = sum_d centers[c,k,d]^2   (one wave per row)
// ---------------------------------------------------------------------------
__global__ __launch_bounds__(256) void k_c2(const float* __restrict__ C,
                                            float* __restrict__ c2) {
  const int t = threadIdx.x, wave = t >> 5, lane = t & 31;
  const int r = blockIdx.x * 8 + wave;
  const float* row = C + (size_t)r * DIM;
  float s = 0.f;
  for (int d = lane; d < DIM; d += 32) { float v = row[d]; s += v * v; }
  s = wave_sum32(s);
  if (lane == 0) c2[r] = s;
}

// ---------------------------------------------------------------------------
// logits = x @ W^T + bias ; idx[b,cb] = argmax over CBS columns.
// Block = (16 rows of B) x (one codebook, 256 cols). 8 waves, each wave owns
// two 16x16 N-tiles, K-loop of 4 via V_WMMA_F32_16X16X4_F32.
// x tile (16x512, 32KB contiguous) is DMA'd into LDS by the TDM.
// ---------------------------------------------------------------------------
__global__ __launch_bounds__(256) void k_logits_argmax(
    const float* __restrict__ x, const float* __restrict__ W,
    const float* __restrict__ bias, int* __restrict__ idxb) {
  __shared__ float As[16 * LDSPITCH];   // x tile, TDM pad layout
  __shared__ float Sc[16 * 256];        // score tile for argmax
  __shared__ float rv[256];
  __shared__ int   ri[256];

  const int t = threadIdx.x;
  const int wave = t >> 5, lane = t & 31, hf = lane >> 4, l16 = lane & 15;
  const int b0 = blockIdx.x * 16;
  const int cb = blockIdx.y;

#ifdef HAVE_TDM
  if (t < 32) {  // wave 0 issues the tensor DMA (TDM is a per-wave op)
    unsigned lds_base = (unsigned)(size_t)(void*)As;
    unsigned long long ga =
        (unsigned long long)(size_t)(const void*)(x + (size_t)b0 * DIM);
    v4u g0;
    g0.x = 1u;                                   // count=1, no gather
    g0.y = lds_base;                             // D#.lds_addr
    g0.z = (unsigned)(ga & 0xffffffffu);         // D#.global_addr[31:0]
    g0.w = (unsigned)((ga >> 32) & 0x01ffffffu)  // D#.global_addr[56:32]
           | (2u << 30);                         // D#.type = 2 (image)
    v8i g1;
    g1[0] = (int)((2u << 16)     // data_size = 4B
                  | (1u << 20)   // pad_enable
                  | (7u << 22)   // pad_interval: every 256 DWORDs
                  | (1u << 25)); // pad_amount: 2 DWORDs
    g1[1] = (int)(512u << 16);   // tensor_dim0 = 512 (bits 79:48, lo16)
    g1[2] = (int)(512u << 16);   // tensor_dim0 hi=0 | tensor_dim1 = 512 lo16
    g1[3] = (int)(512u << 16);   // tensor_dim1 hi=0 | tile_dim0 = 512
    g1[4] = 16;                  // tile_dim1 = 16, tile_dim2 = 0
    g1[5] = 512;                 // tensor_dim0_stride = 512 (lo32)
    g1[6] = 0; g1[7] = 0;        // stride hi, tensor_dim1_stride = 0 (2D)
    v4i gz = {0, 0, 0, 0};       // groups 2/3 unused for 2D tile
    v8i gz8 = {0, 0, 0, 0, 0, 0, 0, 0};  // trailing group (6-arg toolchain)
    __builtin_amdgcn_tensor_load_to_lds(g0, g1, gz, gz, gz8, 0);
    __builtin_amdgcn_s_wait_tensorcnt(0);  // only wave0's TENSORcnt is live
  }
  __syncthreads();               // publish LDS tile to the other 7 waves
#else
  {  // fallback: manual staging with the same split-pad layout
    int r = t >> 4, c16 = t & 15;
    const float* src = x + (size_t)(b0 + r) * DIM;
#pragma unroll
    for (int j = 0; j < DIM / 16; ++j) {
      int d = c16 + j * 16;
      As[r * LDSPITCH + d + ((d >> 8) << 1)] = src[d];
    }
  }
  __syncthreads();
#endif

  const int n0 = wave * 32;
  // B frag (4x16): lane<16 holds (K=k0+v, N=l16); lanes>=16 K=k0+2+v.
  // B[k][n] = W[n][k] -> read 2 consecutive floats of row n.
  const float* w0 = W + ((size_t)cb * CBS + n0 + l16) * DIM + 2 * hf;
  const float* w1 = w0 + (size_t)16 * DIM;

  v8f acc0 = {}; v8f acc1 = {};
  gemm_core_16x32(As, w0, w1, l16, hf, acc0, acc1);

  // D layout: VGPR i holds (M = i + 8*hf, N = l16). LOGITS_SCALE>0 -> dropped.
  const float bs0 = bias[cb * CBS + n0 + l16];
  const float bs1 = bias[cb * CBS + n0 + 16 + l16];
#pragma unroll
  for (int i = 0; i < 8; ++i) {
    int m = i + 8 * hf;
    Sc[m * 256 + n0 + l16]      = acc0[i] + bs0;
    Sc[m * 256 + n0 + 16 + l16] = acc1[i] + bs1;
  }
  __syncthreads();

  // per-row argmax, first-index tie-break (matches jnp.argmax)
  {
    int row = t >> 4, seg = t & 15;
    float best = -INFINITY; int bi = seg * 16;
#pragma unroll
    for (int j = 0; j < 16; ++j) {
      int c = seg * 16 + j;
      float v = Sc[row * 256 + c];
      if (v > best) { best = v; bi = c; }
    }
    rv[t] = best; ri[t] = bi;
  }
  __syncthreads();
  if (t < 16) {
    float best = -INFINITY; int bi = 0;
    for (int s = 0; s < 16; ++s) {          // ascending seg keeps lowest index
      float v = rv[t * 16 + s];
      if (v > best) { best = v; bi = ri[t * 16 + s]; }
    }
    idxb[(b0 + t) * NCB + cb] = bi;
  }
}

// ---------------------------------------------------------------------------
// x_err[b,:] = sum_c centers[c, idx[b,c], :] - x[b,:]
// ---------------------------------------------------------------------------
__global__ __launch_bounds__(256) void k_xerr(
    const float* __restrict__ x, const float* __restrict__ C,
    const int* __restrict__ idxb, float* __restrict__ xerr) {
  __shared__ int ir[NCB];
  const int b = blockIdx.x, t = threadIdx.x;
  if (t < NCB) ir[t] = idxb[b * NCB + t];
  __syncthreads();
  float s0 = -x[(size_t)b * DIM + t];
  float s1 = -x[(size_t)b * DIM + 256 + t];
#pragma unroll
  for (int c = 0; c < NCB; ++c) {
    const float* row = C + ((size_t)c * CBS + ir[c]) * DIM;
    s0 += row[t];
    s1 += row[t + 256];
  }
  xerr[(size_t)b * DIM + t]       = s0;
  xerr[(size_t)b * DIM + 256 + t] = s1;
}

// ---------------------------------------------------------------------------
// proposed[b,cb]: base = x_err - cur_center (built in LDS), WMMA GEMM against
// centers[cb], score = -(c2 + 2*cross)  (base2 is row-constant -> dropped),
// mask current index, argmax.
// ---------------------------------------------------------------------------
__global__ __launch_bounds__(256) void k_propose(
    const float* __restrict__ xerr, const float* __restrict__ C,
    const float* __restrict__ c2b, const int* __restrict__ idxb,
    int* __restrict__ propb) {
  __shared__ float As[16 * LDSPITCH];
  __shared__ float Sc[16 * 256];
  __shared__ float rv[256];
  __shared__ int   ri[256];
  __shared__ int   cur[16];

  const int t = threadIdx.x;
  const int wave = t >> 5, lane = t & 31, hf = lane >> 4, l16 = lane & 15;
  const int b0 = blockIdx.x * 16;
  const int cb = blockIdx.y;

  if (t < 16) cur[t] = idxb[(b0 + t) * NCB + cb];
  __syncthreads();

  // base tile = x_err row - gathered current center row (split-pad layout)
  {
    int r = t >> 4, c16 = t & 15;
    const float* xs = xerr + (size_t)(b0 + r) * DIM;
    const float* ws = C + ((size_t)cb * CBS + cur[r]) * DIM;
#pragma unroll
    for (int j = 0; j < DIM / 16; ++j) {
      int d = c16 + j * 16;
      As[r * LDSPITCH + d + ((d >> 8) << 1)] = xs[d] - ws[d];
    }
  }
  __syncthreads();

  const int n0 = wave * 32;
  const float* w0 = C + ((size_t)cb * CBS + n0 + l16) * DIM + 2 * hf;
  const float* w1 = w0 + (size_t)16 * DIM;

  v8f acc0 = {}; v8f acc1 = {};
  gemm_core_16x32(As, w0, w1, l16, hf, acc0, acc1);

  const float cc0 = c2b[cb * CBS + n0 + l16];
  const float cc1 = c2b[cb * CBS + n0 + 16 + l16];
#pragma unroll
  for (int i = 0; i < 8; ++i) {
    int m = i + 8 * hf;
    Sc[m * 256 + n0 + l16]      = -(cc0 + 2.f * acc0[i]);
    Sc[m * 256 + n0 + 16 + l16] = -(cc1 + 2.f * acc1[i]);
  }
  __syncthreads();

  {
    int row = t >> 4, seg = t & 15;
    const int forbid = cur[row];
    float best = -INFINITY; int bi = seg * 16;
#pragma unroll
    for (int j = 0; j < 16; ++j) {
      int c = seg * 16 + j;
      float v = (c == forbid) ? -INFINITY : Sc[row * 256 + c];
      if (v > best) { best = v; bi = c; }
    }
    rv[t] = best; ri[t] = bi;
  }
  __syncthreads();
  if (t < 16) {
    float best = -INFINITY; int bi = 0;
    for (int s = 0; s < 16; ++s) {
      float v = rv[t * 16 + s];
      if (v > best) { best = v; bi = ri[t * 16 + s]; }
    }
    propb[(b0 + t) * NCB + cb] = bi;
  }
}

// ---------------------------------------------------------------------------
// Subset selection via 8x8 Gram matrix:
//  err(p) = ||xe||^2 + sum_{c in S} 2<xe,dl_c> + sum_{c,c' in S} <dl_c,dl_c'>
// where S = {c : bit c of p == 0}  (== perm row p of the reference).
// One block per b; 256 threads = 256 candidate patterns; argmin p (first min).
// ---------------------------------------------------------------------------
__global__ __launch_bounds__(256) void k_select(
    const float* __restrict__ xerr, const float* __restrict__ C,
    const int* __restrict__ propb, int* __restrict__ idxb) {
  __shared__ float xe[DIM];
  __shared__ float dl[NCB][DIM];
  __shared__ float G[NCB][NCB];
  __shared__ float g0[NCB];
  __shared__ float E0s;
  __shared__ float re[256];
  __shared__ int   rp[256];
  __shared__ int   curI[NCB], propI[NCB];

  const int b = blockIdx.x, t = threadIdx.x;
  const int wave = t >> 5, lane = t & 31;

  if (t < NCB) { curI[t] = idxb[b * NCB + t]; propI[t] = propb[b * NCB + t]; }
  xe[t]       = xerr[(size_t)b * DIM + t];
  xe[t + 256] = xerr[(size_t)b * DIM + 256 + t];
  __syncthreads();

  {  // deltas: wave c builds dl[c][:]
    int c = t >> 5;
    const float* pr = C + ((size_t)c * CBS + propI[c]) * DIM;
    const float* cr = C + ((size_t)c * CBS + curI[c]) * DIM;
    for (int d = lane; d < DIM; d += 32) dl[c][d] = pr[d] - cr[d];
  }
  __syncthreads();

  // 45 dot products of length 512: g0[0..7], E0, G pairs (i<=j)
  for (int task = wave; task < 45; task += 8) {
    const float *u, *v; int gi = -1, gj = -1;
    if (task < 8)        { u = xe; v = dl[task]; }
    else if (task == 8)  { u = xe; v = xe; }
    else {
      int q = task - 9, i = 0;
      while (q >= NCB - i) { q -= NCB - i; ++i; }
      gi = i; gj = i + q;
      u = dl[gi]; v = dl[gj];
    }
    float s = 0.f;
    for (int d = lane; d < DIM; d += 32) s += u[d] * v[d];
    s = wave_sum32(s);
    if (lane == 0) {
      if (task < 8)       g0[task] = s;
      else if (task == 8) E0s = s;
      else                { G[gi][gj] = s; G[gj][gi] = s; }
    }
  }
  __syncthreads();

  {  // thread t evaluates pattern p = t
    const int p = t;
    float e = E0s;
#pragma unroll
    for (int c = 0; c < NCB; ++c) {
      if (((p >> c) & 1) == 0) {
        e += 2.f * g0[c];
#pragma unroll
        for (int c2i = 0; c2i < NCB; ++c2i)
          if (((p >> c2i) & 1) == 0) e += G[c][c2i];
      }
    }
    re[t] = e; rp[t] = p;
  }
  __syncthreads();
  for (int s = 128; s > 0; s >>= 1) {   // argmin, first-index tie-break
    if (t < s) {
      float eo = re[t + s]; int po = rp[t + s];
      if (eo < re[t] || (eo == re[t] && po < rp[t])) { re[t] = eo; rp[t] = po; }
    }
    __syncthreads();
  }
  if (t == 0) {
    int p = rp[0];
#pragma unroll
    for (int c = 0; c < NCB; ++c) {
      bool sel = ((p >> c) & 1) == 0;     // perm[p,c] == 1
      idxb[b * NCB + c] = sel ? propI[c] : curI[c];
    }
  }
}

__global__ __launch_bounds__(256) void k_out(const int* __restrict__ idxb,
                                             float* __restrict__ out, int n) {
  int i = blockIdx.x * blockDim.x + threadIdx.x;
  if (i < n) out[i] = (float)idxb[i];
}

// ---------------------------------------------------------------------------
extern "C" void kernel_launch(void* const* d_in, const int* in_sizes, int n_in,
                              void* d_out, int out_size, void* d_ws, size_t ws_size,
                              hipStream_t stream) {
  const float* x  = (const float*)d_in[0];   // (512, 512)
  const float* wl = (const float*)d_in[1];   // (2048, 512) logits weight
  const float* bl = (const float*)d_in[2];   // (2048,)     logits bias
  const float* to = (const float*)d_in[3];   // (2048, 512) centers
  float* out = (float*)d_out;                // (512, 8) indexes as float

  // workspace: idx(16KB) | prop(16KB) | c2(8KB) | xerr(1MB)
  int*   idxb  = (int*)d_ws;
  int*   propb = idxb + BSZ * NCB;
  float* c2b   = (float*)(propb + BSZ * NCB);
  float* xerr  = c2b + NCB * CBS;

  dim3 blk(256);
  k_c2<<<dim3(NCB * CBS / 8), blk, 0, stream>>>(to, c2b);
  k_logits_argmax<<<dim3(BSZ / 16, NCB), blk, 0, stream>>>(x, wl, bl, idxb);
  for (int it = 0; it < 2; ++it) {
    k_xerr<<<dim3(BSZ), blk, 0, stream>>>(x, to, idxb, xerr);
    k_propose<<<dim3(BSZ / 16, NCB), blk, 0, stream>>>(xerr, to, c2b, idxb, propb);
    k_select<<<dim3(BSZ), blk, 0, stream>>>(xerr, to, propb, idxb);
  }
  k_out<<<dim3((BSZ * NCB + 255) / 256), blk, 0, stream>>>(idxb, out, BSZ * NCB);
}